// P_CustomLayer_77378130804963
// MI455X (gfx1250) — compile-verified
//
#include <hip/hip_runtime.h>

// ---------------------------------------------------------------------------
// DoG-of-LAB, fused per image on MI455X (gfx1250).
//  * 1 workgroup (256 thr = 8 wave32) per 128x128x3 image; grid = 512
//  * whole image (planar LAB) + two blur temps resident in 320 KB WGP LDS
//    -> HBM traffic is the streaming minimum (~100 MB in + ~100 MB out)
//  * separable Gaussian blur = banded 128x128 GEMM on V_WMMA_F32_16X16X4_F32
//    (f32 in / f32 acc, matches reference precision). Bands: sigma=0.64 -> 6
//    fixed K-chunks of 4, sigma=2.98 -> 10 fixed K-chunks (constant-trip,
//    fully unrolled; out-of-band chunk entries are exact zeros).
//  * all tile/band indices routed through readfirstlane so loop control is
//    scalar and EXEC stays all-ones around WMMA (ISA 7.12 requirement).
// ---------------------------------------------------------------------------

typedef __attribute__((ext_vector_type(2))) float v2f;
typedef __attribute__((ext_vector_type(8))) float v8f;

#define HW    128
#define PIX   (HW * HW)          // 16384 floats per plane (64 KB)
#define NPIXC (PIX * 3)

struct F3 { float x, y, z; };    // 12-byte packed pixel

__device__ __forceinline__ v8f wmma_f32_step(v2f a, v2f b, v8f c) {
  // D(16x16,f32) = A(16x4,f32) * B(4x16,f32) + C
  return __builtin_amdgcn_wmma_f32_16x16x4_f32(
      false, a, false, b, (short)0, c, false, false);
}

__device__ __forceinline__ float srgb_lin(float x) {
  return x > 0.04045f ? powf((x + 0.055f) * (1.0f / 1.055f), 2.4f)
                      : x * (1.0f / 12.92f);
}

__device__ __forceinline__ float lab_f(float t) {
  return t > 0.008856f ? cbrtf(t) : 7.787f * t + (16.0f / 116.0f);
}

// Build the two 128x128 banded blur matrices (reflect padding folded in).
// A[0]: sigma=0.64 (r=3), A[1]: sigma=2.98 (r=12).  1 block x 256 threads.
__global__ void build_blur_mats(float* __restrict__ A) {
  const int t     = threadIdx.x;
  const int which = t >> 7;
  const int i     = t & 127;
  const float sigma = which ? 2.98f : 0.64f;
  const int   r     = which ? 12 : 3;        // int(4*sigma + 0.5)
  float* row = A + which * PIX + i * HW;
  for (int j = 0; j < HW; ++j) row[j] = 0.0f;
  float norm = 0.0f;
  for (int tap = -r; tap <= r; ++tap) {
    const float x = (float)tap / sigma;
    norm += expf(-0.5f * x * x);
  }
  const float inv = 1.0f / norm;
  for (int tap = -r; tap <= r; ++tap) {
    const float x = (float)tap / sigma;
    const float w = expf(-0.5f * x * x) * inv;
    int j = i + tap;                         // symmetric reflect
    if (j < 0)      j = -j - 1;
    if (j > HW - 1) j = 2 * HW - 1 - j;
    row[j] += w;
  }
}

__device__ __forceinline__ int clamp_s(int v, int lo, int hi) {
  return v < lo ? lo : (v > hi ? hi : v);
}

__global__ __launch_bounds__(256, 1)
void dog_fused(const float* __restrict__ xin, float* __restrict__ out,
               const float* __restrict__ Amats) {
  extern __shared__ float smem[];            // 5 planes * 64 KB = 320 KB
  // planes 0..2: LAB channels (later overwritten in place by their DoG)
  float* t1 = smem + 3 * PIX;                // H-blur sigma1 temp
  float* t2 = smem + 4 * PIX;                // H-blur sigma2 temp
  const float* A1 = Amats;
  const float* A2 = Amats + PIX;

  const int tid    = threadIdx.x;
  const int lane   = tid & 31;
  const int half   = lane >> 4;              // 0: lanes 0-15, 1: lanes 16-31
  const int l16    = lane & 15;
  // scalarize the wave id so all tile/band math is SGPR-resident
  const int wave_s = __builtin_amdgcn_readfirstlane(tid >> 5);

  const float* src = xin + (size_t)blockIdx.x * NPIXC;
  float*       dst = out + (size_t)blockIdx.x * NPIXC;

  // ---- Phase 1: sRGB -> CIELAB, planar into LDS (12B packed loads) ----
  for (int p = tid; p < PIX; p += 256) {
    const F3 v = ((const F3*)src)[p];
    const float lr = srgb_lin(v.x), lg = srgb_lin(v.y), lb = srgb_lin(v.z);
    const float X = 0.412453f * lr + 0.357580f * lg + 0.180423f * lb;
    const float Y = 0.212671f * lr + 0.715160f * lg + 0.072169f * lb;
    const float Z = 0.019334f * lr + 0.119193f * lg + 0.950227f * lb;
    const float fx = lab_f(X * (1.0f / 0.95047f));
    const float fy = lab_f(Y);
    const float fz = lab_f(Z * (1.0f / 1.08883f));
    smem[p]           = 116.0f * fy - 16.0f;     // L plane
    smem[PIX + p]     = 500.0f * (fx - fy);      // a plane
    smem[2 * PIX + p] = 200.0f * (fy - fz);      // b plane
  }
  __syncthreads();

  // ---- Per-channel banded GEMM blurs; DoG written back into own plane ----
  for (int c = 0; c < 3; ++c) {
    float* labc = smem + c * PIX;

    // H blur: t1 = A1 @ labc ; t2 = A2 @ labc
#pragma unroll 1
    for (int it = 0; it < 8; ++it) {
      const int tile = wave_s + 8 * it;      // scalar
      const int mt = tile >> 3, nt = tile & 7;
      const int arow = mt * 16 + l16;
      const int ncol = nt * 16 + l16;
      {
        const int kcs = clamp_s(4 * mt - 1, 0, 26);       // scalar, 6 chunks
        const float* Ab = A1 + arow * HW + kcs * 4 + 2 * half;
        const float* Bb = labc + (kcs * 4 + 2 * half) * HW + ncol;
        v8f acc = {};
#pragma unroll
        for (int u = 0; u < 6; ++u) {
          const v2f a = *(const v2f*)(Ab + 4 * u);
          v2f b; b.x = Bb[4 * u * HW]; b.y = Bb[(4 * u + 1) * HW];
          acc = wmma_f32_step(a, b, acc);
        }
#pragma unroll
        for (int j = 0; j < 8; ++j)
          t1[(mt * 16 + j + 8 * half) * HW + ncol] = acc[j];
      }
      {
        const int kcs = clamp_s(4 * mt - 3, 0, 22);       // scalar, 10 chunks
        const float* Ab = A2 + arow * HW + kcs * 4 + 2 * half;
        const float* Bb = labc + (kcs * 4 + 2 * half) * HW + ncol;
        v8f acc = {};
#pragma unroll
        for (int u = 0; u < 10; ++u) {
          const v2f a = *(const v2f*)(Ab + 4 * u);
          v2f b; b.x = Bb[4 * u * HW]; b.y = Bb[(4 * u + 1) * HW];
          acc = wmma_f32_step(a, b, acc);
        }
#pragma unroll
        for (int j = 0; j < 8; ++j)
          t2[(mt * 16 + j + 8 * half) * HW + ncol] = acc[j];
      }
    }
    __syncthreads();

    // W blur + DoG: labc = t1 @ A1^T - t2 @ A2^T
#pragma unroll 1
    for (int it = 0; it < 8; ++it) {
      const int tile = wave_s + 8 * it;      // scalar
      const int mt = tile >> 3, nt = tile & 7;
      const int arow = mt * 16 + l16;
      const int ncol = nt * 16 + l16;
      v8f acc1 = {};
      {
        const int kcs = clamp_s(4 * nt - 1, 0, 26);       // scalar, 6 chunks
        const float* Tb = t1 + arow * HW + kcs * 4 + 2 * half;
        const float* Wb = A1 + ncol * HW + kcs * 4 + 2 * half;
#pragma unroll
        for (int u = 0; u < 6; ++u) {
          const v2f a = *(const v2f*)(Tb + 4 * u);        // LDS b64
          const v2f b = *(const v2f*)(Wb + 4 * u);        // global b64
          acc1 = wmma_f32_step(a, b, acc1);
        }
      }
      v8f acc2 = {};
      {
        const int kcs = clamp_s(4 * nt - 3, 0, 22);       // scalar, 10 chunks
        const float* Tb = t2 + arow * HW + kcs * 4 + 2 * half;
        const float* Wb = A2 + ncol * HW + kcs * 4 + 2 * half;
#pragma unroll
        for (int u = 0; u < 10; ++u) {
          const v2f a = *(const v2f*)(Tb + 4 * u);
          const v2f b = *(const v2f*)(Wb + 4 * u);
          acc2 = wmma_f32_step(a, b, acc2);
        }
      }
#pragma unroll
      for (int j = 0; j < 8; ++j) {
        const int m = mt * 16 + j + 8 * half;
        labc[m * HW + ncol] = acc1[j] - acc2[j];          // raw DoG, in place
      }
    }
    __syncthreads();
  }

  // ---- Per-image min/max of DoG-L (t1 plane is free -> scratch) ----
  float mn = 3.402823466e38f, mx = -3.402823466e38f;
  for (int p = tid; p < PIX; p += 256) {
    const float v = smem[p];
    mn = fminf(mn, v);
    mx = fmaxf(mx, v);
  }
  float* red = t1;
  red[tid]       = mn;
  red[256 + tid] = mx;
  __syncthreads();
  for (int s = 128; s > 0; s >>= 1) {
    if (tid < s) {
      red[tid]       = fminf(red[tid], red[tid + s]);
      red[256 + tid] = fmaxf(red[256 + tid], red[256 + tid + s]);
    }
    __syncthreads();
  }
  const float lmin = red[0];
  const float linv = 1.0f / (red[256] - lmin);

  // ---- Final packed store: all 3 channels per pixel in one 12B store ----
  for (int p = tid; p < PIX; p += 256) {
    F3 o;
    o.x = (smem[p] - lmin) * linv;
    o.y = (smem[PIX + p] + 128.0f) * (1.0f / 255.0f);
    o.z = (smem[2 * PIX + p] * 0.1f + 128.0f) * (1.0f / 255.0f);
    ((F3*)dst)[p] = o;
  }
}

extern "C" void kernel_launch(void* const* d_in, const int* in_sizes, int n_in,
                              void* d_out, int out_size, void* d_ws, size_t ws_size,
                              hipStream_t stream) {
  (void)in_sizes; (void)n_in; (void)out_size; (void)ws_size;
  const float* xin = (const float*)d_in[0];
  float*       out = (float*)d_out;
  float*       Amats = (float*)d_ws;         // [2][128][128] f32 = 128 KB

  build_blur_mats<<<1, 256, 0, stream>>>(Amats);
  dog_fused<<<512, 256, 5 * PIX * sizeof(float), stream>>>(xin, out, Amats);
}